// SurvivalNLLLoss_11510512353859
// MI455X (gfx1250) — compile-verified
//
#include <hip/hip_runtime.h>
#include <hip/hip_bf16.h>

// SurvivalNLLLoss for MI455X (gfx1250, wave32).
// Memory-bound streaming kernel: 134MB read once -> ~5.8us floor @ 23.3TB/s.
// Transcendental pressure reduced 8x via masked chunk-products:
//   sum_{i<=t} log(1-h_i) == sum_chunks log( prod_{i in chunk, i<=t} (1-h_i) )
// (8 factors, each >= 0.01 -> product >= 1e-16, safely inside f32 range).

#define EPS_F 1e-7f

typedef float v2f __attribute__((ext_vector_type(2)));
typedef float v8f __attribute__((ext_vector_type(8)));

__device__ __forceinline__ float clampf(float h) {
    return fminf(fmaxf(h, EPS_F), 1.0f - EPS_F);
}

// Kernel 1: each thread owns 2 adjacent columns (float2 loads).
// Masked product per 8-row chunk, one logf per chunk; h[t] captured via cndmask.
// Deterministic LDS tree reduce -> one partial per block.
__global__ void __launch_bounds__(256)
survival_nll_main(const float* __restrict__ hz,
                  const int* __restrict__ et,
                  const int* __restrict__ cen,
                  float* __restrict__ partial,
                  int T, int B) {
    const int tx   = threadIdx.x;
    const int tid  = blockIdx.x * blockDim.x + tx;
    const int col0 = tid * 2;
    const int B2   = B >> 1;           // columns in float2 units (B is even)

    float thread_loss = 0.0f;

    if (col0 + 1 < B) {
        int t0 = et[col0];     t0 = min(max(t0, 0), T - 1);
        int t1 = et[col0 + 1]; t1 = min(max(t1, 0), T - 1);
        const bool c0 = (cen[col0] != 0);
        const bool c1 = (cen[col0 + 1] != 0);

        const v2f* __restrict__ hz2 = (const v2f*)hz;
        const size_t cbase = (size_t)(col0 >> 1);

        float acc0 = 0.0f, acc1 = 0.0f;
        float ht0 = 0.5f, ht1 = 0.5f;   // captured h at event row

        int i0 = 0;
        // Main unrolled loop: 8 rows per iteration, batched loads for MLP.
        for (; i0 + 8 <= T; i0 += 8) {
            // Stream-ahead prefetch (global_prefetch_b8 on gfx1250).
            if (i0 + 24 < T) {
                __builtin_prefetch(&hz2[(size_t)(i0 + 24) * B2 + cbase], 0, 1);
            }
            v2f v[8];
#pragma unroll
            for (int k = 0; k < 8; ++k) {
                v[k] = hz2[(size_t)(i0 + k) * B2 + cbase];
            }
            float p0 = 1.0f, p1 = 1.0f;   // masked survival products (chunk)
#pragma unroll
            for (int k = 0; k < 8; ++k) {
                const int i = i0 + k;
                float h0 = clampf(v[k].x);
                float h1 = clampf(v[k].y);
                p0 *= (i <= t0) ? (1.0f - h0) : 1.0f;
                p1 *= (i <= t1) ? (1.0f - h1) : 1.0f;
                ht0 = (i == t0) ? h0 : ht0;    // v_cndmask capture, no reload
                ht1 = (i == t1) ? h1 : ht1;
            }
            acc0 += logf(p0);
            acc1 += logf(p1);
        }
        // Remainder rows (T not multiple of 8).
        if (i0 < T) {
            float p0 = 1.0f, p1 = 1.0f;
            for (; i0 < T; ++i0) {
                v2f v = hz2[(size_t)i0 * B2 + cbase];
                float h0 = clampf(v.x);
                float h1 = clampf(v.y);
                p0 *= (i0 <= t0) ? (1.0f - h0) : 1.0f;
                p1 *= (i0 <= t1) ? (1.0f - h1) : 1.0f;
                ht0 = (i0 == t0) ? h0 : ht0;
                ht1 = (i0 == t1) ? h1 : ht1;
            }
            acc0 += logf(p0);
            acc1 += logf(p1);
        }

        // Epilogue: 2 extra logs per column, zero extra global traffic.
        // loss_cens  = -acc ; loss_event = -acc + log(1-ht) - log(ht)
        float fix0 = logf(1.0f - ht0) - logf(ht0);
        float fix1 = logf(1.0f - ht1) - logf(ht1);
        float loss0 = -acc0 + (c0 ? 0.0f : fix0);
        float loss1 = -acc1 + (c1 ? 0.0f : fix1);
        thread_loss = loss0 + loss1;
    }

    // Deterministic block reduction.
    __shared__ float red[256];
    red[tx] = thread_loss;
    __syncthreads();
#pragma unroll
    for (int s = 128; s > 0; s >>= 1) {
        if (tx < s) red[tx] += red[tx + s];
        __syncthreads();
    }
    if (tx == 0) partial[blockIdx.x] = red[0];
}

// Kernel 2: one wave32 reduces the per-block partials.
// Uses V_WMMA_F32_16X16X4_F32 with A = ones(16x4) so D[m,n] = colsum_n of B.
// Since every B VGPR slot maps bijectively onto a matrix element, summing
// VGPR0 of D over one 16-lane half yields the exact total sum of all B slots,
// independent of the precise slot->element mapping.
__global__ void __launch_bounds__(32)
survival_nll_reduce(const float* __restrict__ partial, int n,
                    float invB, float* __restrict__ out) {
    const int lane = threadIdx.x;   // 32 lanes, EXEC all ones at WMMA
    float s = 0.0f;
    for (int i = lane; i < n; i += 32) s += partial[i];   // fixed order

    v2f a; a.x = 1.0f; a.y = 1.0f;       // A = all-ones 16x4
    v2f b; b.x = s;    b.y = 0.0f;       // B holds the 32 lane partials
    v8f c = {};
    v8f d = __builtin_amdgcn_wmma_f32_16x16x4_f32(
        /*neg_a=*/false, a, /*neg_b=*/false, b,
        /*c_mod=*/(short)0, c, /*reuse_a=*/false, /*reuse_b=*/false);

    float x = d[0];                       // lane L (0..15): colsum_L
    // Sum within each 16-lane half; lanes 0..15 hold 16 distinct colsums.
    x += __shfl_xor(x, 1, 16);
    x += __shfl_xor(x, 2, 16);
    x += __shfl_xor(x, 4, 16);
    x += __shfl_xor(x, 8, 16);

    if (lane == 0) out[0] = x * invB;
}

extern "C" void kernel_launch(void* const* d_in, const int* in_sizes, int n_in,
                              void* d_out, int out_size, void* d_ws, size_t ws_size,
                              hipStream_t stream) {
    const float* hz  = (const float*)d_in[0];   // hazard  [T, B] f32
    const int*   et  = (const int*)d_in[1];     // event_times [B]
    const int*   cen = (const int*)d_in[2];     // censored [B]

    const int B = in_sizes[1];
    const int T = in_sizes[0] / B;

    float* out     = (float*)d_out;
    float* partial = (float*)d_ws;              // nblocks floats of scratch

    const int threads      = 256;
    const int colsPerBlock = threads * 2;
    const int nblocks      = (B + colsPerBlock - 1) / colsPerBlock;

    survival_nll_main<<<nblocks, threads, 0, stream>>>(hz, et, cen, partial, T, B);

    const float invB = (float)(1.0 / (double)B);
    survival_nll_reduce<<<1, 32, 0, stream>>>(partial, nblocks, invB, out);
}